// HMM_2980707303629
// MI455X (gfx1250) — compile-verified
//
#include <hip/hip_runtime.h>
#include <math.h>

#define N_STATES 64
#define N_OBS    10000
#define BATCH    256
#define TMAXLEN  2048
#define PITCH    68          // padded LDS row pitch (floats) -> conflict-free strided reads

typedef __attribute__((ext_vector_type(2))) float v2f;
typedef __attribute__((ext_vector_type(8))) float v8f;

// workspace layout (float offsets)
#define WS_PT     0          // [64][64]  PT[k][i] = P(i | k) = softmax_i(A[:,k])
#define WS_LOGPI  4096       // [64]
#define WS_LSE    4160       // [64]      row logsumexp of emission logits
#define WS_LET    4224       // [10000][64] log_E^T[obs][n] (normalized, contiguous per obs)

// ---------------- prep: column softmax of transition logits -> dense probs PT[k][i]
__global__ __launch_bounds__(64)
void hmm_prep_trans(const float* __restrict__ A, float* __restrict__ W) {
    __shared__ float red[64];
    const int k = blockIdx.x, i = threadIdx.x;
    float v = A[i * N_STATES + k];
    red[i] = v; __syncthreads();
    for (int s = 32; s > 0; s >>= 1) { if (i < s) red[i] = fmaxf(red[i], red[i + s]); __syncthreads(); }
    float mx = red[0]; __syncthreads();
    float e = expf(v - mx);
    red[i] = e; __syncthreads();
    for (int s = 32; s > 0; s >>= 1) { if (i < s) red[i] += red[i + s]; __syncthreads(); }
    W[WS_PT + k * 64 + i] = e / red[0];
}

// ---------------- prep: log_softmax of pi
__global__ __launch_bounds__(64)
void hmm_prep_pi(const float* __restrict__ pi, float* __restrict__ W) {
    __shared__ float red[64];
    const int i = threadIdx.x;
    float v = pi[i];
    red[i] = v; __syncthreads();
    for (int s = 32; s > 0; s >>= 1) { if (i < s) red[i] = fmaxf(red[i], red[i + s]); __syncthreads(); }
    float mx = red[0]; __syncthreads();
    red[i] = expf(v - mx); __syncthreads();
    for (int s = 32; s > 0; s >>= 1) { if (i < s) red[i] += red[i + s]; __syncthreads(); }
    W[WS_LOGPI + i] = v - (mx + logf(red[0]));
}

// ---------------- prep: per-state logsumexp over 10000 emission logits
__global__ __launch_bounds__(256)
void hmm_prep_emis_lse(const float* __restrict__ E, float* __restrict__ W) {
    __shared__ float red[256];
    const int n = blockIdx.x, t = threadIdx.x;
    float mx = -INFINITY;
    for (int o = t; o < N_OBS; o += 256) mx = fmaxf(mx, E[n * N_OBS + o]);
    red[t] = mx; __syncthreads();
    for (int s = 128; s > 0; s >>= 1) { if (t < s) red[t] = fmaxf(red[t], red[t + s]); __syncthreads(); }
    mx = red[0]; __syncthreads();
    float sum = 0.f;
    for (int o = t; o < N_OBS; o += 256) sum += expf(E[n * N_OBS + o] - mx);
    red[t] = sum; __syncthreads();
    for (int s = 128; s > 0; s >>= 1) { if (t < s) red[t] += red[t + s]; __syncthreads(); }
    if (t == 0) W[WS_LSE + n] = mx + logf(red[0]);
}

// ---------------- prep: normalized transposed emissions  LET[o][n] = E[n][o] - LSE[n]
__global__ __launch_bounds__(256)
void hmm_prep_let(const float* __restrict__ E, float* __restrict__ W) {
    const int idx = blockIdx.x * blockDim.x + threadIdx.x;
    if (idx >= N_OBS * N_STATES) return;
    const int n = idx / N_OBS;       // consecutive threads sweep o -> coalesced reads of E
    const int o = idx % N_OBS;
    W[WS_LET + o * 64 + n] = E[idx] - W[WS_LSE + n];
}

// ---------------- main: latency-optimized forward recursion with fp32 WMMA chains
__global__ __launch_bounds__(128)
void hmm_forward(const int* __restrict__ x, const int* __restrict__ Tlen,
                 const float* __restrict__ W, float* __restrict__ out) {
    __shared__ float alpha[16 * PITCH];
    __shared__ float pbuf[16 * PITCH];
    __shared__ float mrow[16];
    __shared__ int   obs[16];
    __shared__ int   Tl[16];

    const float* PT  = W + WS_PT;
    const float* LPI = W + WS_LOGPI;
    const float* LET = W + WS_LET;

    const int tid  = threadIdx.x;
    const int b0   = blockIdx.x * 16;
    const int lane = tid & 31;
    const int wid  = tid >> 5;          // wave 0..3, owns output columns [16w,16w+16)
    const int ln   = lane & 15;
    const int koff = (lane >> 4) * 2;   // 0 or 2 (K half select, f32 WMMA A/B layout)
    const int col0 = wid * 16;

    // Constant B fragments of PT (B[k,i] = P(i|k)), resident in VGPRs for entire sequence.
    v2f bfrag[16];
#pragma unroll
    for (int kc = 0; kc < 16; ++kc) {
        const int kb = 4 * kc + koff;
        v2f b;
        b.x = PT[kb * 64 + col0 + ln];
        b.y = PT[(kb + 1) * 64 + col0 + ln];
        bfrag[kc] = b;
    }

    const int r  = tid >> 3;            // batch row handled in phase 1 (0..15)
    const int cg = tid & 7;             // 8-lane group within row

    if (tid < 16) {
        Tl[tid]  = Tlen[b0 + tid];
        obs[tid] = x[(b0 + tid) * TMAXLEN + 0];
    }
    __syncthreads();

    int tmx = 0;
#pragma unroll
    for (int q = 0; q < 16; ++q) tmx = max(tmx, Tl[q]);

    // alpha0 = log_E[:, x0] + log_pi
    {
        const int o = obs[r];
#pragma unroll
        for (int j = 0; j < 8; ++j) {
            const int c = cg * 8 + j;
            alpha[r * PITCH + c] = LET[o * 64 + c] + LPI[c];
        }
    }
    __syncthreads();

    if (tid < 16 && Tl[tid] == 1) {
        float m2 = -INFINITY;
        for (int c = 0; c < 64; ++c) m2 = fmaxf(m2, alpha[tid * PITCH + c]);
        float s = 0.f;
        for (int c = 0; c < 64; ++c) s += expf(alpha[tid * PITCH + c] - m2);
        out[b0 + tid] = m2 + logf(s);
    }

    for (int t = 1; t < tmx; ++t) {
        // ---- phase 1: row max + exp into pbuf (p = exp(alpha - m)), fetch obs_t
        if (tid < 16) obs[tid] = x[(b0 + tid) * TMAXLEN + t];
        float v[8];
        float mx = -INFINITY;
        const int base = r * PITCH + cg * 8;
#pragma unroll
        for (int j = 0; j < 8; ++j) { v[j] = alpha[base + j]; mx = fmaxf(mx, v[j]); }
        mx = fmaxf(mx, __shfl_xor(mx, 1));
        mx = fmaxf(mx, __shfl_xor(mx, 2));
        mx = fmaxf(mx, __shfl_xor(mx, 4));
#pragma unroll
        for (int j = 0; j < 8; ++j) pbuf[base + j] = expf(v[j] - mx);
        if (cg == 0) mrow[r] = mx;
        __syncthreads();

        // ---- phase 2: new_p[16x16 tile] = p[16x64] * PT[64x16] via 16 chained fp32 WMMAs
        v8f acc = {};
#pragma unroll
        for (int kc = 0; kc < 16; ++kc) {
            v2f a = *(const v2f*)&pbuf[ln * PITCH + 4 * kc + koff];
            acc = __builtin_amdgcn_wmma_f32_16x16x4_f32(
                /*neg_a=*/false, a, /*neg_b=*/false, bfrag[kc],
                /*c_mod=*/(short)0, acc, /*reuse_a=*/false, /*reuse_b=*/false);
        }
        // ---- writeback: alpha' = m + log(new_p) + log_E[:, x_t]
        const int col   = col0 + ln;
        const int rbase = (lane < 16) ? 0 : 8;
#pragma unroll
        for (int j = 0; j < 8; ++j) {
            const int row = rbase + j;
            const float na = mrow[row] + logf(acc[j]) + LET[obs[row] * 64 + col];
            alpha[row * PITCH + col] = na;
        }
        __syncthreads();

        // ---- termination: emit logsumexp for rows ending at this step
        if (tid < 16 && (Tl[tid] - 1) == t) {
            float m2 = -INFINITY;
            for (int c = 0; c < 64; ++c) m2 = fmaxf(m2, alpha[tid * PITCH + c]);
            float s = 0.f;
            for (int c = 0; c < 64; ++c) s += expf(alpha[tid * PITCH + c] - m2);
            out[b0 + tid] = m2 + logf(s);
        }
    }
}

extern "C" void kernel_launch(void* const* d_in, const int* in_sizes, int n_in,
                              void* d_out, int out_size, void* d_ws, size_t ws_size,
                              hipStream_t stream) {
    const int*   x  = (const int*)d_in[0];
    const int*   T  = (const int*)d_in[1];
    const float* pi = (const float*)d_in[2];
    const float* A  = (const float*)d_in[3];
    const float* E  = (const float*)d_in[4];
    float* out = (float*)d_out;
    float* W   = (float*)d_ws;   // needs ~2.6 MB

    hipLaunchKernelGGL(hmm_prep_trans,    dim3(64),   dim3(64),  0, stream, A, W);
    hipLaunchKernelGGL(hmm_prep_pi,       dim3(1),    dim3(64),  0, stream, pi, W);
    hipLaunchKernelGGL(hmm_prep_emis_lse, dim3(64),   dim3(256), 0, stream, E, W);
    hipLaunchKernelGGL(hmm_prep_let,      dim3((N_OBS * N_STATES + 255) / 256), dim3(256), 0, stream, E, W);
    hipLaunchKernelGGL(hmm_forward,       dim3(BATCH / 16), dim3(128), 0, stream, x, T, W, out);
}